// AdditiveAttention_62654982914792
// MI455X (gfx1250) — compile-verified
//
#include <hip/hip_runtime.h>
#include <hip/hip_bf16.h>
#include <math.h>

typedef __attribute__((ext_vector_type(2))) float v2f;
typedef __attribute__((ext_vector_type(8))) float v8f;

// gfx1250 has a hardware V_TANH_F32 (TRANS pipe, co-executes with VALU).
// The DEVICE pass must resolve to the single-instruction builtin; #error
// otherwise so the compile log tells us instead of silently inlining the
// ocml polynomial. Host pass just needs something that parses.
#if defined(__HIP_DEVICE_COMPILE__)
#  if __has_builtin(__builtin_amdgcn_tanhf)
#    define TANHF(x) __builtin_amdgcn_tanhf(x)
#  elif __has_builtin(__builtin_amdgcn_tanh_f32)
#    define TANHF(x) __builtin_amdgcn_tanh_f32(x)
#  else
#    error "no hardware tanh builtin on this toolchain for gfx1250 device pass"
#  endif
#else
#  define TANHF(x) tanhf(x)   /* host pass only, never executed */
#endif

#define BH 16   // b*h
#define T  384
#define S  384
#define D  64

// ---------------------------------------------------------------------------
// Kernel 1: projections  qp = q @ Wq^T,  kp = k @ Wk^T   (WMMA f32 16x16x4)
// One GEMM each: M = BH*T = 6144 rows, K = N = 64.
// 256-thread block = 8 waves, one 16x16 C-tile per wave along m.
// A-frag: lane l%16 = row M, K = 2*(lane/16)+j           [ISA 7.12.2]
// B-frag: lane l%16 = col N, K = 2*(lane/16)+j ; B(k,n) = W[n*64 + k]
// C/D:    VGPR r, lanes 0-15 -> M=r, lanes 16-31 -> M=r+8
// ---------------------------------------------------------------------------
__global__ void proj_wmma_kernel(const float* __restrict__ q,
                                 const float* __restrict__ k,
                                 const float* __restrict__ Wq,
                                 const float* __restrict__ Wk,
                                 float* __restrict__ ws) {
  const int wid   = threadIdx.x >> 5;
  const int mtile = blockIdx.x * 8 + wid;   // 0..383
  const int ntile = blockIdx.y;             // 0..3
  const int which = blockIdx.z;             // 0 = q-proj, 1 = k-proj
  const float* __restrict__ A = which ? k : q;
  const float* __restrict__ W = which ? Wk : Wq;
  float* __restrict__ out = ws + (size_t)which * (BH * T * D);

  const int lane = threadIdx.x & 31;
  const int half = lane >> 4;               // K-group: 0 -> K=0..1, 1 -> K=2..3
  const int l    = lane & 15;
  const int m0   = mtile * 16;
  const int n0   = ntile * 16;

  v8f c = {};
  #pragma unroll
  for (int k0 = 0; k0 < D; k0 += 4) {
    const int kk = k0 + 2 * half;
    v2f a; a.x = A[(m0 + l) * D + kk];  a.y = A[(m0 + l) * D + kk + 1];
    v2f b; b.x = W[(n0 + l) * D + kk];  b.y = W[(n0 + l) * D + kk + 1];
    c = __builtin_amdgcn_wmma_f32_16x16x4_f32(false, a, false, b,
                                              (short)0, c, false, false);
  }
  #pragma unroll
  for (int r = 0; r < 8; ++r)
    out[(m0 + r + 8 * half) * D + n0 + l] = c[r];
}

// ---------------------------------------------------------------------------
// Kernel 2 (the hot one, trans-bound): score + softmax.
// One block per (bh, t) row, 384 threads (12 waves), thread s owns column s.
//   score = sum_d v[d] * tanh(qp[t,d] + kp[s,d])  -> 64 v_tanh_f32 / thread
// 4 independent accumulators break the FMA dependency chain.
// Softmax: wave32 shuffle reductions + 12-entry LDS combine.
// ---------------------------------------------------------------------------
__global__ void score_softmax_kernel(const float* __restrict__ ws,
                                     const float* __restrict__ vw,
                                     float* __restrict__ attn) {
  const int t  = blockIdx.x;
  const int bh = blockIdx.y;
  const int s  = threadIdx.x;               // 0..383
  const float* __restrict__ qp = ws;
  const float* __restrict__ kp = ws + (size_t)BH * T * D;

  __shared__ float sq[D];
  __shared__ float sv[D];
  __shared__ float wred[12];

  if (threadIdx.x < D) {
    sq[threadIdx.x] = qp[((size_t)bh * T + t) * D + threadIdx.x];
    sv[threadIdx.x] = vw[threadIdx.x];
  }
  __syncthreads();

  const float4* __restrict__ kr =
      (const float4*)(kp + ((size_t)bh * S + s) * D);
  float a0 = 0.f, a1 = 0.f, a2 = 0.f, a3 = 0.f;
  #pragma unroll
  for (int d4 = 0; d4 < D / 4; ++d4) {
    const float4 kv = kr[d4];
    a0 += sv[4 * d4 + 0] * TANHF(sq[4 * d4 + 0] + kv.x);
    a1 += sv[4 * d4 + 1] * TANHF(sq[4 * d4 + 1] + kv.y);
    a2 += sv[4 * d4 + 2] * TANHF(sq[4 * d4 + 2] + kv.z);
    a3 += sv[4 * d4 + 3] * TANHF(sq[4 * d4 + 3] + kv.w);
  }
  const float acc = (a0 + a1) + (a2 + a3);

  const int lane = threadIdx.x & 31;
  const int wid  = threadIdx.x >> 5;

  // block max
  float m = acc;
  #pragma unroll
  for (int off = 16; off; off >>= 1) m = fmaxf(m, __shfl_xor(m, off));
  if (lane == 0) wred[wid] = m;
  __syncthreads();
  float bm = wred[0];
  #pragma unroll
  for (int i = 1; i < 12; ++i) bm = fmaxf(bm, wred[i]);

  const float e = __expf(acc - bm);
  __syncthreads();                          // all reads of wred (max) done

  // block sum
  float ssum = e;
  #pragma unroll
  for (int off = 16; off; off >>= 1) ssum += __shfl_xor(ssum, off);
  if (lane == 0) wred[wid] = ssum;
  __syncthreads();
  float bs = 0.f;
  #pragma unroll
  for (int i = 0; i < 12; ++i) bs += wred[i];

  attn[((size_t)bh * T + t) * S + s] = e / bs;
}

// ---------------------------------------------------------------------------
// Kernel 3: out = attn @ value per bh slab  (WMMA f32 16x16x4)
// 256-thread block = 8 waves; the 8 waves share one value n-column (near-cache
// reuse of B fragments). K = 384 stepped by 4 (96 wmma / tile).
// B(k,n) = value[k*64 + n].
// ---------------------------------------------------------------------------
__global__ void av_wmma_kernel(const float* __restrict__ attn,
                               const float* __restrict__ value,
                               float* __restrict__ out) {
  const int wid   = threadIdx.x >> 5;
  const int mtile = blockIdx.x * 8 + wid;   // 0..23
  const int ntile = blockIdx.y;             // 0..3
  const int bh    = blockIdx.z;             // 0..15
  const float* __restrict__ A = attn  + (size_t)bh * T * S;
  const float* __restrict__ B = value + (size_t)bh * S * D;

  const int lane = threadIdx.x & 31;
  const int half = lane >> 4;
  const int l    = lane & 15;
  const int m0   = mtile * 16;
  const int n0   = ntile * 16;

  v8f c = {};
  #pragma unroll 8
  for (int k0 = 0; k0 < S; k0 += 4) {
    const int kk = k0 + 2 * half;
    v2f a; a.x = A[(m0 + l) * S + kk];   a.y = A[(m0 + l) * S + kk + 1];
    v2f b; b.x = B[kk * D + n0 + l];     b.y = B[(kk + 1) * D + n0 + l];
    c = __builtin_amdgcn_wmma_f32_16x16x4_f32(false, a, false, b,
                                              (short)0, c, false, false);
  }
  #pragma unroll
  for (int r = 0; r < 8; ++r)
    out[((size_t)bh * T + m0 + r + 8 * half) * D + n0 + l] = c[r];
}

// ---------------------------------------------------------------------------
extern "C" void kernel_launch(void* const* d_in, const int* in_sizes, int n_in,
                              void* d_out, int out_size, void* d_ws, size_t ws_size,
                              hipStream_t stream) {
  (void)in_sizes; (void)n_in; (void)out_size; (void)ws_size;
  const float* q     = (const float*)d_in[0];
  const float* key   = (const float*)d_in[1];
  const float* value = (const float*)d_in[2];
  const float* Wq    = (const float*)d_in[3];
  const float* Wk    = (const float*)d_in[4];
  const float* vw    = (const float*)d_in[5];

  float* out  = (float*)d_out;                       // (b,h,t,d)
  float* attn = out + (size_t)BH * T * D;            // (b,h,t,s), 2nd output
  float* ws   = (float*)d_ws;                        // qp (1.5MB) + kp (1.5MB)

  // 1) projections into workspace: ws[0..]=qp, ws[BH*T*D..]=kp
  proj_wmma_kernel<<<dim3(BH * T / 16 / 8, D / 16, 2), 256, 0, stream>>>(
      q, key, Wq, Wk, ws);

  // 2) additive score + softmax -> attn (directly into d_out's 2nd region)
  score_softmax_kernel<<<dim3(T, BH), S, 0, stream>>>(ws, vw, attn);

  // 3) out = attn @ value
  av_wmma_kernel<<<dim3(T / 16 / 8, D / 16, BH), 256, 0, stream>>>(
      attn, value, out);
}